// DeformableCorrelation_45664092291268
// MI455X (gfx1250) — compile-verified
//
#include <hip/hip_runtime.h>

typedef __attribute__((ext_vector_type(2))) float v2f;
typedef __attribute__((ext_vector_type(8))) float v8f;

#define BB   8
#define CC   256
#define HH   128
#define WW   128
#define HWW  (HH * WW)

// One wave -> 16 consecutive output pixels (b, y, xb..xb+15).
// Per 4-channel step: load A (feat1, 16x4) and four gathered B operands
// (feat2 bilinear corners, 4x16), do 4x V_WMMA_F32_16X16X4_F32.
// diag(acc_corner) after 64 steps = the 16 per-pixel channel dots.
__global__ __launch_bounds__(256) void deform_corr_wmma_kernel(
    const float* __restrict__ feat1,
    const float* __restrict__ feat2,
    const float* __restrict__ offset,
    float* __restrict__ out)
{
    const int lane = threadIdx.x & 31;
    const int wIdx = threadIdx.x >> 5;          // wave in block (0..7)
    const int half = lane >> 4;                 // 0: lanes 0-15, 1: lanes 16-31
    const int n    = lane & 15;                 // pixel slot within tile

    const int wave = blockIdx.x * 8 + wIdx;     // 0..8191 tiles
    const int b    = wave >> 10;                // 1024 tiles per batch (128 rows * 8 xtiles)
    const int rem  = wave & 1023;
    const int y    = rem >> 3;
    const int xb   = (rem & 7) << 4;
    const int x    = xb + n;

    // ---- per-pixel sampling parameters (lanes 16-31 mirror lanes 0-15) ----
    const float dy = offset[((b * 2 + 0) * HH + y) * WW + x];
    const float dx = offset[((b * 2 + 1) * HH + y) * WW + x];
    const float py = (float)y + dy;
    const float px = (float)x + dx;
    const float fy = floorf(py);
    const float fx = floorf(px);
    const float wy = py - fy;
    const float wx = px - fx;
    const int y0 = (int)fy, x0 = (int)fx;
    const int y1 = y0 + 1,  x1 = x0 + 1;

    const float v00 = (y0 >= 0 && y0 < HH && x0 >= 0 && x0 < WW) ? 1.0f : 0.0f;
    const float v01 = (y0 >= 0 && y0 < HH && x1 >= 0 && x1 < WW) ? 1.0f : 0.0f;
    const float v10 = (y1 >= 0 && y1 < HH && x0 >= 0 && x0 < WW) ? 1.0f : 0.0f;
    const float v11 = (y1 >= 0 && y1 < HH && x1 >= 0 && x1 < WW) ? 1.0f : 0.0f;

    const int y0c = min(max(y0, 0), HH - 1);
    const int y1c = min(max(y1, 0), HH - 1);
    const int x0c = min(max(x0, 0), WW - 1);
    const int x1c = min(max(x1, 0), WW - 1);

    const unsigned s00 = (unsigned)(y0c * WW + x0c);
    const unsigned s01 = (unsigned)(y0c * WW + x1c);
    const unsigned s10 = (unsigned)(y1c * WW + x0c);
    const unsigned s11 = (unsigned)(y1c * WW + x1c);

    const float* __restrict__ f1p = feat1 + (size_t)b * CC * HWW + (unsigned)(y * WW + x);
    const float* __restrict__ f2p = feat2 + (size_t)b * CC * HWW;

    v8f a00 = {}, a01 = {}, a10 = {}, a11 = {};

    for (int c = 0; c < CC; c += 4) {
        // Both A and B use lane-half-splits-K: this lane supplies K = 2*half, 2*half+1
        const unsigned ch = (unsigned)(c + 2 * half) * HWW;

        v2f A;
        A[0] = f1p[ch];
        A[1] = f1p[ch + HWW];

        v2f B00, B01, B10, B11;
        B00[0] = f2p[ch + s00];        B00[1] = f2p[ch + HWW + s00];
        B01[0] = f2p[ch + s01];        B01[1] = f2p[ch + HWW + s01];
        B10[0] = f2p[ch + s10];        B10[1] = f2p[ch + HWW + s10];
        B11[0] = f2p[ch + s11];        B11[1] = f2p[ch + HWW + s11];

        a00 = __builtin_amdgcn_wmma_f32_16x16x4_f32(false, A, false, B00, (short)0, a00, false, false);
        a01 = __builtin_amdgcn_wmma_f32_16x16x4_f32(false, A, false, B01, (short)0, a01, false, false);
        a10 = __builtin_amdgcn_wmma_f32_16x16x4_f32(false, A, false, B10, (short)0, a10, false, false);
        a11 = __builtin_amdgcn_wmma_f32_16x16x4_f32(false, A, false, B11, (short)0, a11, false, false);
    }

    // ---- extract diag(acc) : pixel n -> (VGPR n, lane n) for n<8,
    //                          (VGPR n-8, lane n+16) => lane r+24 holds pixel r+8 ----
    __shared__ float red[8][16][4];
#pragma unroll
    for (int r = 0; r < 8; ++r) {
        if (lane == r) {
            red[wIdx][r][0] = a00[r];
            red[wIdx][r][1] = a01[r];
            red[wIdx][r][2] = a10[r];
            red[wIdx][r][3] = a11[r];
        }
        if (lane == r + 24) {
            red[wIdx][r + 8][0] = a00[r];
            red[wIdx][r + 8][1] = a01[r];
            red[wIdx][r + 8][2] = a10[r];
            red[wIdx][r + 8][3] = a11[r];
        }
    }
    __syncthreads();

    if (lane < 16) {
        const float d00 = red[wIdx][n][0];
        const float d01 = red[wIdx][n][1];
        const float d10 = red[wIdx][n][2];
        const float d11 = red[wIdx][n][3];
        const float w00 = (1.0f - wy) * (1.0f - wx) * v00;
        const float w01 = (1.0f - wy) * wx * v01;
        const float w10 = wy * (1.0f - wx) * v10;
        const float w11 = wy * wx * v11;
        out[(b * HH + y) * WW + x] = w00 * d00 + w01 * d01 + w10 * d10 + w11 * d11;
    }
}

extern "C" void kernel_launch(void* const* d_in, const int* in_sizes, int n_in,
                              void* d_out, int out_size, void* d_ws, size_t ws_size,
                              hipStream_t stream) {
    (void)in_sizes; (void)n_in; (void)d_ws; (void)ws_size; (void)out_size;
    const float* feat1  = (const float*)d_in[0];
    const float* feat2  = (const float*)d_in[1];
    const float* offset = (const float*)d_in[2];
    float* out = (float*)d_out;

    // 8192 wave-tiles (8 batches * 128 rows * 8 x-tiles), 8 waves per 256-thread block
    dim3 grid(1024), block(256);
    deform_corr_wmma_kernel<<<grid, block, 0, stream>>>(feat1, feat2, offset, out);
}